// ConditionalEdgeMasker_25924422599238
// MI455X (gfx1250) — compile-verified
//
#include <hip/hip_runtime.h>

// gfx1250 (MI455X) — memory-bound elementwise edge-mask kernel.
// Per edge: 64B feature record (4 x b128 NT loads) + 4B base load + 4B NT store.
// Physics scalars are block-uniform -> scalar loads (s_load_b32, KMcnt path).

typedef float v4f __attribute__((ext_vector_type(4)));

#define MIN_JUMP_ENERGY        0.5f
#define MIN_VELOCITY_THRESHOLD 0.1f
#define MIN_WALL_JUMP_SPEED    1.0f
#define PHYS_THR               0.5f

__global__ __launch_bounds__(256) void edge_mask_kernel(
    const float* __restrict__ feat,   // [B, E, 16]
    const float* __restrict__ phys,   // [B, 18]
    const float* __restrict__ base,   // [B, E]
    float*       __restrict__ out,    // [B, E]
    int E)
{
    const int b = blockIdx.y;
    const int e = blockIdx.x * blockDim.x + threadIdx.x;
    if (e >= E) return;

    // Block-uniform physics scalars -> compiler emits s_load_b32 (SMEM/KMcnt).
    const float* __restrict__ p = phys + (size_t)b * 18;
    const float vel = p[2];    // vel_magnitude
    const float wc  = p[5];    // wall_contact
    const float ke  = p[9];    // kinetic_energy
    const float cj  = p[16];   // can_jump
    const float cwj = p[17];   // can_wall_jump

    // 64B feature record per edge: 4 x global_load_b128 with NT hint
    // (streamed once over 204.8 MB -> do not pollute the 192 MB L2).
    const v4f* __restrict__ fp = (const v4f*)(feat + ((size_t)b * E + e) * 16);
    const v4f f0 = __builtin_nontemporal_load(fp + 0);  // feat 0..3
    const v4f f1 = __builtin_nontemporal_load(fp + 1);  // feat 4..7
    const v4f f2 = __builtin_nontemporal_load(fp + 2);  // feat 8..11
    const v4f f3 = __builtin_nontemporal_load(fp + 3);  // feat 12..15

    // argmax over feat[0:6], first-occurrence semantics (strict >).
    float v0 = f0.x, v1 = f0.y, v2 = f0.z, v3 = f0.w, v4 = f1.x, v5 = f1.y;
    int   et   = 0;
    float best = v0;
    if (v1 > best) { best = v1; et = 1; }
    if (v2 > best) { best = v2; et = 2; }
    if (v3 > best) { best = v3; et = 3; }
    if (v4 > best) { best = v4; et = 4; }
    if (v5 > best) { best = v5; et = 5; }

    const float energy_cost  = f2.z;  // feat[10]
    const float min_velocity = f3.x;  // feat[12]
    const float max_velocity = f3.y;  // feat[13]
    const float req_jump     = f3.z;  // feat[14]
    const float req_wall     = f3.w;  // feat[15]

    // Conditions, evaluated with jnp.select first-true-wins ordering.
    const bool c1 = (et == 1) && (cj < PHYS_THR);                       // EDGE_TYPE_JUMP, cannot jump
    const bool c2 = (et == 3);                                          // EDGE_TYPE_WALL_SLIDE
    const bool d2 = (wc < PHYS_THR) || (vel < MIN_VELOCITY_THRESHOLD);
    const bool c3 = (req_jump > PHYS_THR);
    const bool d3 = (cj < PHYS_THR) || (vel < min_velocity);
    const bool c4 = (req_wall > PHYS_THR);
    //   python precedence: wc<thr | (cwj<thr & vel<MIN_WALL_JUMP_SPEED)
    const bool d4 = (wc < PHYS_THR) || ((cwj < PHYS_THR) && (vel < MIN_WALL_JUMP_SPEED));
    const bool c5 = (et == 1);
    const bool d5 = (ke < energy_cost * MIN_JUMP_ENERGY);

    bool disable = c1 ? true
                 : c2 ? d2
                 : c3 ? d3
                 : c4 ? d4
                 : c5 ? d5
                 : false;

    const bool over  = (max_velocity > 0.0f) && (vel > max_velocity);
    const bool under = (!over) && (vel < min_velocity);
    disable = disable || over || under;

    const size_t idx = (size_t)b * E + e;
    const float bm = base[idx];
    __builtin_nontemporal_store(bm * (1.0f - (disable ? 1.0f : 0.0f)), out + idx);
}

extern "C" void kernel_launch(void* const* d_in, const int* in_sizes, int n_in,
                              void* d_out, int out_size, void* d_ws, size_t ws_size,
                              hipStream_t stream) {
    const float* feat = (const float*)d_in[0];  // [B, E, 16]
    const float* phys = (const float*)d_in[1];  // [B, 18]
    const float* base = (const float*)d_in[2];  // [B, E]
    float*       out  = (float*)d_out;          // [B, E]

    const int B = in_sizes[1] / 18;             // 32
    const int E = in_sizes[2] / B;              // 100000

    const int block = 256;                      // 8 wave32 waves
    dim3 grid((E + block - 1) / block, B);
    edge_mask_kernel<<<grid, dim3(block), 0, stream>>>(feat, phys, base, out, E);
}